// MMDetRoIPaster_72232759984541
// MI455X (gfx1250) — compile-verified
//
#include <hip/hip_runtime.h>

// ---- problem constants (from the reference) ----
#define N_ROIS 128
#define R      28          // mask resolution
#define IH     800
#define IW     800
#define TILES_X (IW / 16)  // 50
#define TILES_Y (IH / 16)  // 50
#define WAVES_PER_BLOCK 8  // 256 threads, wave32

typedef __attribute__((ext_vector_type(2))) float v2f;
typedef __attribute__((ext_vector_type(8))) float v8f;

// out[n] = Wy[n] * mask[n] * Wx[n]^T  done as:
//   A (per wave, once per 16-row strip) = row-bilinear-interp of mask -> 16x28 f32
//   per 16-col tile: B = column selection weights (28x16), D += A*B via
//   7x V_WMMA_F32_16X16X4_F32 (K = 28 = 7 chunks of 4).
__global__ __launch_bounds__(256) void roi_paste_wmma(
    const float* __restrict__ masks,   // [128, 1, 28, 28]
    const float* __restrict__ boxes,   // [128, 4] xyxy
    float* __restrict__ out)           // [128, 800, 800]
{
    __shared__ float smask[R * R];     // 3136 B

    const int n    = blockIdx.x / TILES_Y;
    const int ty   = blockIdx.x % TILES_Y;
    const int tid  = threadIdx.x;
    const int lane = tid & 31;
    const int wave = tid >> 5;

    // ---- stage this RoI's mask into LDS ----
    const float* gm = masks + (size_t)n * (R * R);
    for (int i = tid; i < R * R; i += 256) smask[i] = gm[i];
    __syncthreads();

    const float bx0 = boxes[n * 4 + 0], by0 = boxes[n * 4 + 1];
    const float bx1 = boxes[n * 4 + 2], by1 = boxes[n * 4 + 3];
    const float sy = (float)R / (by1 - by0);
    const float sx = (float)R / (bx1 - bx0);

    // ---- A-matrix build: 16x4 f32 WMMA layout ----
    // lane 0-15: M = lane, holds K = 4kc+0 (vgpr0) / 4kc+1 (vgpr1)
    // lane 16-31: M = lane-16, holds K = 4kc+2 / 4kc+3
    const int m     = lane & 15;
    const int khalf = (lane >> 4) * 2;

    const int   y  = ty * 16 + m;
    const float py = ((float)y + 0.5f - by0) * sy - 0.5f;   // input-pixel y coord
    const float yf = floorf(py);
    const float wy = py - yf;
    const int   yi = (int)yf;
    const float w0 = (yi     >= 0 && yi     < R) ? (1.0f - wy) : 0.0f;
    const float w1 = (yi + 1 >= 0 && yi + 1 < R) ? wy          : 0.0f;
    const int   y0c = min(max(yi,     0), R - 1);
    const int   y1c = min(max(yi + 1, 0), R - 1);
    const float* r0 = smask + y0c * R;
    const float* r1 = smask + y1c * R;

    v2f A[7];
    #pragma unroll
    for (int kc = 0; kc < 7; ++kc) {
        const int c0 = 4 * kc + khalf;                       // even -> 8B aligned
        const float2 a0 = *reinterpret_cast<const float2*>(r0 + c0); // ds_load_b64
        const float2 a1 = *reinterpret_cast<const float2*>(r1 + c0);
        A[kc].x = w0 * a0.x + w1 * a1.x;
        A[kc].y = w0 * a0.y + w1 * a1.y;
    }

    // ---- sweep x-tiles of this 16-row strip; 8 waves round-robin ----
    for (int tx = wave; tx < TILES_X; tx += WAVES_PER_BLOCK) {
        const int   x  = tx * 16 + m;                        // N = m
        const float px = ((float)x + 0.5f - bx0) * sx - 0.5f;
        const float xf = floorf(px);
        const float wx = px - xf;
        const int   xi = (int)xf;

        // B[c][j] = (c==xi)?(1-wx) : (c==xi+1)?wx : 0.
        // Zero padding is implicit: an out-of-range neighbor index never
        // matches any c in [0,28), so it contributes nothing.
        v8f acc = {0.f, 0.f, 0.f, 0.f, 0.f, 0.f, 0.f, 0.f};
        #pragma unroll
        for (int kc = 0; kc < 7; ++kc) {
            const int c0 = 4 * kc + khalf;
            v2f B;
            B.x = (c0     == xi) ? (1.0f - wx) : ((c0     == xi + 1) ? wx : 0.0f);
            B.y = (c0 + 1 == xi) ? (1.0f - wx) : ((c0 + 1 == xi + 1) ? wx : 0.0f);
            acc = __builtin_amdgcn_wmma_f32_16x16x4_f32(
                false, A[kc], false, B, (short)0, acc, false, false);
        }

        // D layout: VGPR v -> row M = v + 8*(lane>=16), col N = lane&15.
        // 16 lanes x 4B = coalesced 64B row segments per store.
        float* o = out + ((size_t)n * IH + (size_t)(ty * 16)) * IW
                       + (size_t)tx * 16 + m;
        const int mbase = (lane >> 4) * 8;
        #pragma unroll
        for (int v = 0; v < 8; ++v) {
            o[(size_t)(mbase + v) * IW] = acc[v];
        }
    }
}

extern "C" void kernel_launch(void* const* d_in, const int* in_sizes, int n_in,
                              void* d_out, int out_size, void* d_ws, size_t ws_size,
                              hipStream_t stream) {
    const float* masks = (const float*)d_in[0];   // [128,1,28,28] f32
    const float* boxes = (const float*)d_in[1];   // [128,4] f32
    float* out = (float*)d_out;                   // [128,800,800] f32
    (void)in_sizes; (void)n_in; (void)out_size; (void)d_ws; (void)ws_size;

    dim3 grid(N_ROIS * TILES_Y);  // 6400 blocks: one 16-row strip per block
    dim3 block(256);              // 8 wave32 waves
    roi_paste_wmma<<<grid, block, 0, stream>>>(masks, boxes, out);
}